// RecurrentEncoder_72997264163398
// MI455X (gfx1250) — compile-verified
//
#include <hip/hip_runtime.h>
#include <math.h>

// ---------------------------------------------------------------------------
// LSTM encoder for MI455X (gfx1250, wave32).
// Persistent per-layer kernel: grid barrier per timestep, cell state in VGPRs,
// all GEMM work on V_WMMA_F32_16X16X4_F32 (fp32 matrix pipe).
// Weights (16MB) stay in L2.
// ---------------------------------------------------------------------------

typedef __attribute__((ext_vector_type(2))) float v2f;
typedef __attribute__((ext_vector_type(8))) float v8f;

#define BATCH   32
#define SEQ     1024
#define HID     512
#define GATEC   2048     // 4*HID
#define NBLK    8        // persistent blocks (must all be co-resident)
#define NTHR    256      // 8 waves per block
#define PAD_ID  0

static __device__ __forceinline__ v8f wmma_f32(v2f a, v2f b, v8f c) {
    // D = A(16x4 f32) x B(4x16 f32) + C(16x16 f32)
    return __builtin_amdgcn_wmma_f32_16x16x4_f32(
        /*neg_a=*/false, a, /*neg_b=*/false, b,
        /*c_mod=*/(short)0, c, /*reuse_a=*/false, /*reuse_b=*/false);
}

static __device__ __forceinline__ float sigm(float x) {
    return 1.0f / (1.0f + __expf(-x));
}

// One timestep's 4 accumulation chains: 4 gates x (x-GEMM [+ h-GEMM]).
// K = 512 -> 128 k-steps of K=4.
template <bool HASH>
static __device__ __forceinline__ void gemm_step(
    const float* __restrict__ xp, const float* __restrict__ hp,
    const float* __restrict__ wi0, const float* __restrict__ wi1,
    const float* __restrict__ wi2, const float* __restrict__ wi3,
    const float* __restrict__ wh0, const float* __restrict__ wh1,
    const float* __restrict__ wh2, const float* __restrict__ wh3,
    v8f& a0, v8f& a1, v8f& a2, v8f& a3)
{
#pragma unroll 4
    for (int k4 = 0; k4 < HID / 4; ++k4) {
        const int ko = k4 * 4;
        v2f xf = *(const v2f*)(xp + ko);
        v2f w0 = *(const v2f*)(wi0 + ko);
        v2f w1 = *(const v2f*)(wi1 + ko);
        v2f w2 = *(const v2f*)(wi2 + ko);
        v2f w3 = *(const v2f*)(wi3 + ko);
        a0 = wmma_f32(xf, w0, a0);
        a1 = wmma_f32(xf, w1, a1);
        a2 = wmma_f32(xf, w2, a2);
        a3 = wmma_f32(xf, w3, a3);
        if (HASH) {
            v2f hv = *(const v2f*)(hp + ko);
            v2f u0 = *(const v2f*)(wh0 + ko);
            v2f u1 = *(const v2f*)(wh1 + ko);
            v2f u2 = *(const v2f*)(wh2 + ko);
            v2f u3 = *(const v2f*)(wh3 + ko);
            a0 = wmma_f32(hv, u0, a0);
            a1 = wmma_f32(hv, u1, a1);
            a2 = wmma_f32(hv, u2, a2);
            a3 = wmma_f32(hv, u3, a3);
        }
    }
}

__global__ void __launch_bounds__(NTHR)
lstm_layer_kernel(const int*   __restrict__ ids,    // [B,S]    (layer0 path)
                  const float* __restrict__ embed,  // [V,512]  (layer0 path; row 0 is zero)
                  const float* __restrict__ xbuf,   // [S,B,512](layer1 path)
                  const float* __restrict__ w_ih,   // [2048,512] row-major
                  const float* __restrict__ w_hh,   // [2048,512] row-major
                  const float* __restrict__ b_ih,   // [2048]
                  const float* __restrict__ b_hh,   // [2048]
                  float*       __restrict__ hout,   // [S,B,512] this layer's h sequence
                  float*       __restrict__ hf,     // [B,512]
                  float*       __restrict__ cf,     // [B,512]
                  unsigned*    __restrict__ bar,    // monotonic barrier counter (pre-zeroed)
                  int layer0)
{
    const int lane  = threadIdx.x & 31;
    const int wave  = threadIdx.x >> 5;          // 0..7
    const int mtile = wave & 1;                  // batch tile: rows [mtile*16, +16)
    const int htile = blockIdx.x * 4 + (wave >> 1);  // 0..31 -> h cols [htile*16, +16)
    const int hbase = htile * 16;
    const int lm    = lane & 15;
    const int ksel  = (lane >> 4) << 1;          // 0 (lanes 0-15) or 2 (lanes 16-31)
    const int mrow  = mtile * 16 + lm;           // A-fragment source row for this lane

    // Per-lane weight row pointers: B-fragment for gate column n = g*512+hbase+lm,
    // element (K,K+1) at W[n*512 + 4*k4 + ksel] (W row-major => this is W^T tile).
    const size_t n0 = (size_t)(0 * HID + hbase + lm) * HID + ksel;
    const size_t n1 = (size_t)(1 * HID + hbase + lm) * HID + ksel;
    const size_t n2 = (size_t)(2 * HID + hbase + lm) * HID + ksel;
    const size_t n3 = (size_t)(3 * HID + hbase + lm) * HID + ksel;
    const float *wi0 = w_ih + n0, *wi1 = w_ih + n1, *wi2 = w_ih + n2, *wi3 = w_ih + n3;
    const float *wh0 = w_hh + n0, *wh1 = w_hh + n1, *wh2 = w_hh + n2, *wh3 = w_hh + n3;

    // Bias (per output column; same for every row in the tile).
    const float bi = b_ih[0 * HID + hbase + lm] + b_hh[0 * HID + hbase + lm];
    const float bf = b_ih[1 * HID + hbase + lm] + b_hh[1 * HID + hbase + lm];
    const float bg = b_ih[2 * HID + hbase + lm] + b_hh[2 * HID + hbase + lm];
    const float bo = b_ih[3 * HID + hbase + lm] + b_hh[3 * HID + hbase + lm];

    // Cell state lives in registers for the entire sequence.
    v8f c_state, h_last;
#pragma unroll
    for (int r = 0; r < 8; ++r) { c_state[r] = 0.0f; h_last[r] = 0.0f; }

    const int mhi = (lane >= 16) ? 8 : 0;        // C/D layout: VGPR r holds rows r / r+8

    for (int t = 0; t < SEQ; ++t) {
        // A-fragment source row pointer for x_t.
        const float* xp;
        if (layer0) {
            const int id = ids[mrow * SEQ + t];  // embed row PAD_ID is all-zero => mask free
            xp = embed + (size_t)id * HID + ksel;
        } else {
            xp = xbuf + ((size_t)t * BATCH + mrow) * HID + ksel;
        }
        const float* hp = hout + ((size_t)(t - 1) * BATCH + mrow) * HID + ksel;

        v8f a0, a1, a2, a3;
#pragma unroll
        for (int r = 0; r < 8; ++r) { a0[r] = 0.f; a1[r] = 0.f; a2[r] = 0.f; a3[r] = 0.f; }

        if (t > 0)
            gemm_step<true >(xp, hp, wi0, wi1, wi2, wi3, wh0, wh1, wh2, wh3, a0, a1, a2, a3);
        else
            gemm_step<false>(xp, hp, wi0, wi1, wi2, wi3, wh0, wh1, wh2, wh3, a0, a1, a2, a3);

        // Activations + cell update; all four gates for (row,col) sit in the same lane.
#pragma unroll
        for (int r = 0; r < 8; ++r) {
            const float ig = sigm(a0[r] + bi);
            const float fg = sigm(a1[r] + bf);
            const float gg = tanhf(a2[r] + bg);
            const float og = sigm(a3[r] + bo);
            const float cn = fg * c_state[r] + ig * gg;
            c_state[r] = cn;
            h_last[r]  = og * tanhf(cn);
        }

        // Store h_t (C/D layout: VGPR r -> row mtile*16 + r + mhi, col hbase+lm).
#pragma unroll
        for (int r = 0; r < 8; ++r) {
            const int m = mtile * 16 + r + mhi;
            hout[((size_t)t * BATCH + m) * HID + hbase + lm] = h_last[r];
        }

        // Grid-wide barrier: h_t must be device-visible before anyone starts t+1.
        __threadfence();
        __syncthreads();
        if (threadIdx.x == 0) {
            __hip_atomic_fetch_add(bar, 1u, __ATOMIC_RELEASE, __HIP_MEMORY_SCOPE_AGENT);
            const unsigned target = (unsigned)(NBLK * (t + 1));
            while (__hip_atomic_load(bar, __ATOMIC_ACQUIRE, __HIP_MEMORY_SCOPE_AGENT) < target)
                __builtin_amdgcn_s_sleep(1);
        }
        __syncthreads();
    }

    // Final hidden / cell state for this layer.
#pragma unroll
    for (int r = 0; r < 8; ++r) {
        const int m = mtile * 16 + r + mhi;
        hf[(size_t)m * HID + hbase + lm] = h_last[r];
        cf[(size_t)m * HID + hbase + lm] = c_state[r];
    }
}

__global__ void init_barriers_kernel(unsigned* bar) {
    if (threadIdx.x < 2) bar[threadIdx.x] = 0u;
}

__global__ void pad_mask_kernel(const int* __restrict__ ids, float* __restrict__ mask) {
    const int i = blockIdx.x * blockDim.x + threadIdx.x;  // i = t*B + b
    if (i < SEQ * BATCH) {
        const int t = i / BATCH, b = i % BATCH;
        mask[i] = (ids[b * SEQ + t] == PAD_ID) ? 1.0f : 0.0f;
    }
}

extern "C" void kernel_launch(void* const* d_in, const int* in_sizes, int n_in,
                              void* d_out, int out_size, void* d_ws, size_t ws_size,
                              hipStream_t stream) {
    (void)in_sizes; (void)n_in; (void)out_size; (void)ws_size;

    const int*   ids   = (const int*)  d_in[0];   // [B,S]
    const float* embed = (const float*)d_in[1];   // [32000,512]
    const float* w_ih  = (const float*)d_in[2];   // [2,2048,512]
    const float* w_hh  = (const float*)d_in[3];   // [2,2048,512]
    const float* b_ih  = (const float*)d_in[4];   // [2,2048]
    const float* b_hh  = (const float*)d_in[5];   // [2,2048]

    // d_out layout: outs [S,B,H] | hf [2,B,H] | cf [2,B,H] | padding_mask [S,B]
    float* outs  = (float*)d_out;
    float* hf    = outs + (size_t)SEQ * BATCH * HID;
    float* cf    = hf   + (size_t)2 * BATCH * HID;
    float* maskp = cf   + (size_t)2 * BATCH * HID;

    // Workspace: [0..1] barrier counters, +256B aligned: X0 = layer-0 h sequence (64MB).
    unsigned* bar = (unsigned*)d_ws;
    float*    X0  = (float*)((char*)d_ws + 256);

    init_barriers_kernel<<<1, 32, 0, stream>>>(bar);
    pad_mask_kernel<<<(SEQ * BATCH + 255) / 256, 256, 0, stream>>>(ids, maskp);

    const size_t woff = (size_t)GATEC * HID;   // per-layer weight stride
    // Layer 0: x = embed[ids] (pad row zero), h-sequence -> X0 (workspace).
    lstm_layer_kernel<<<NBLK, NTHR, 0, stream>>>(
        ids, embed, X0 /*unused*/, w_ih, w_hh, b_ih, b_hh,
        X0, hf, cf, bar + 0, /*layer0=*/1);
    // Layer 1: x = X0[t], h-sequence -> outs (== final output sequence).
    lstm_layer_kernel<<<NBLK, NTHR, 0, stream>>>(
        ids, embed, X0, w_ih + woff, w_hh + woff, b_ih + GATEC, b_hh + GATEC,
        outs, hf + (size_t)BATCH * HID, cf + (size_t)BATCH * HID, bar + 1, /*layer0=*/0);
}